// Model_78417512890408
// MI455X (gfx1250) — compile-verified
//
#include <hip/hip_runtime.h>
#include <stdint.h>

// Problem constants (from the reference)
#define L_SEQ   96
#define CH      7
#define K_FEAT  7328          // L + L(L+1)/2 + sum over (i<=j<7) of (L-j)
#define PRED    720
#define BATCH   256
#define M_ROWS  (BATCH * CH)  // 1792
#define NTILES  3             // 3 x 16 = 48 N-cols per wave; 720 = 15 * 48
#define MTILES  2             // 2 x 16 = 32 M-rows per wave
#define KSPLIT  2
#define KHALF   (K_FEAT / KSPLIT)   // 3664 = 229 * 16

typedef float v2f __attribute__((ext_vector_type(2)));
typedef float v8f __attribute__((ext_vector_type(8)));

// ---------------------------------------------------------------------------
// Kernel 1: build the monomial index table.
// table[s] = t0 | (t1<<8) | (t2<<16); 255 means "unused" (multiply by 1).
// ---------------------------------------------------------------------------
__global__ void build_table_kernel(uint32_t* __restrict__ table) {
    int s = blockIdx.x * blockDim.x + threadIdx.x;
    if (s >= K_FEAT) return;
    uint32_t t0 = 0, t1 = 255, t2 = 255;
    if (s < L_SEQ) {
        t0 = (uint32_t)s;
    } else {
        int u = s - L_SEQ;
        const int quad_total = L_SEQ * (L_SEQ + 1) / 2;  // 4656
        if (u < quad_total) {
            int i = 0;
            while (u >= (L_SEQ - i)) { u -= (L_SEQ - i); ++i; }
            t0 = (uint32_t)(i + u);
            t1 = (uint32_t)i;
        } else {
            u -= quad_total;
            bool found = false;
            for (int i = 0; i < CH && !found; ++i) {
                for (int j = i; j < CH; ++j) {
                    int len = L_SEQ - j;
                    if (u < len) { t0 = (uint32_t)(j + u); t1 = (uint32_t)i; t2 = (uint32_t)j; found = true; break; }
                    u -= len;
                }
            }
        }
    }
    table[s] = t0 | (t1 << 8) | (t2 << 16);
}

// ---------------------------------------------------------------------------
// Kernel 2: materialize A[m][s] (m = b*7 + c) with the 96-float series in LDS.
// ---------------------------------------------------------------------------
__global__ void expand_feat_kernel(const float* __restrict__ x,
                                   const uint32_t* __restrict__ table,
                                   float* __restrict__ A) {
    __shared__ float xr[L_SEQ];
    const int m = blockIdx.y;
    const int b = m / CH, c = m % CH;
    for (int t = threadIdx.x; t < L_SEQ; t += blockDim.x)
        xr[t] = x[((size_t)b * L_SEQ + t) * CH + c];
    __syncthreads();
    const int s = blockIdx.x * blockDim.x + threadIdx.x;
    if (s < K_FEAT) {
        uint32_t e = table[s];
        int t0 = (int)(e & 255u), t1 = (int)((e >> 8) & 255u), t2 = (int)((e >> 16) & 255u);
        float v = xr[t0];
        if (t1 != 255) v *= xr[t1];
        if (t2 != 255) v *= xr[t2];
        A[(size_t)m * K_FEAT + s] = v;
    }
}

// ---------------------------------------------------------------------------
// Kernel 3: fp32 WMMA GEMM, K split in 2, partial sums to workspace.
// Wave tile: 32(M) x 48(N) -> 6 accumulators; block = 8 waves -> 256 x 48.
// Grid (15, 7, 2); every dimension divides exactly -> EXEC all-1s for WMMA.
// Per K-step of 4: 5 x global_load_b64 feed 6 x v_wmma_f32_16x16x4_f32.
// ---------------------------------------------------------------------------
__global__ __launch_bounds__(256) void wmma_gemm_kernel(
    const float* __restrict__ A, const float* __restrict__ W,
    float* __restrict__ partial) {

    const int lane = threadIdx.x & 31;
    const int wave = threadIdx.x >> 5;
    const int m0   = (blockIdx.y * 8 + wave) * (16 * MTILES);
    const int n0   = blockIdx.x * (16 * NTILES);
    const int kb   = blockIdx.z * KHALF;

    const int khalf = (lane >> 4) * 2;  // lanes 0-15 -> K+0, lanes 16-31 -> K+2

    const float* Arow0 = A + (size_t)(m0 + (lane & 15)) * K_FEAT + kb + khalf;
    const float* Arow1 = Arow0 + (size_t)16 * K_FEAT;
    const float* Wc0   = W + (size_t)(n0 + 0 * 16 + (lane & 15)) * K_FEAT + kb + khalf;
    const float* Wc1   = W + (size_t)(n0 + 1 * 16 + (lane & 15)) * K_FEAT + kb + khalf;
    const float* Wc2   = W + (size_t)(n0 + 2 * 16 + (lane & 15)) * K_FEAT + kb + khalf;

    v8f acc00 = {}, acc01 = {}, acc02 = {};
    v8f acc10 = {}, acc11 = {}, acc12 = {};

    // KHALF = 3664 = 229 * 16 ; 4 WMMA K-steps per outer iteration.
    for (int k = 0; k < KHALF; k += 16) {
        __builtin_prefetch(Arow0 + k + 1024, 0, 1);
        __builtin_prefetch(Arow1 + k + 1024, 0, 1);
#pragma unroll
        for (int kk = 0; kk < 16; kk += 4) {
            v2f a0 = *(const v2f*)(Arow0 + k + kk);
            v2f a1 = *(const v2f*)(Arow1 + k + kk);
            v2f b0 = *(const v2f*)(Wc0 + k + kk);
            v2f b1 = *(const v2f*)(Wc1 + k + kk);
            v2f b2 = *(const v2f*)(Wc2 + k + kk);
            acc00 = __builtin_amdgcn_wmma_f32_16x16x4_f32(false, a0, false, b0, (short)0, acc00, false, false);
            acc10 = __builtin_amdgcn_wmma_f32_16x16x4_f32(false, a1, false, b0, (short)0, acc10, false, false);
            acc01 = __builtin_amdgcn_wmma_f32_16x16x4_f32(false, a0, false, b1, (short)0, acc01, false, false);
            acc11 = __builtin_amdgcn_wmma_f32_16x16x4_f32(false, a1, false, b1, (short)0, acc11, false, false);
            acc02 = __builtin_amdgcn_wmma_f32_16x16x4_f32(false, a0, false, b2, (short)0, acc02, false, false);
            acc12 = __builtin_amdgcn_wmma_f32_16x16x4_f32(false, a1, false, b2, (short)0, acc12, false, false);
        }
    }

    // Store partial sums: P[z][m * 720 + p]. C/D layout: VGPR v -> M = v (+8 for
    // lanes 16-31); N = lane & 15.
    float* P = partial + (size_t)blockIdx.z * M_ROWS * PRED;
    const int mbase = m0 + ((lane >> 4) ? 8 : 0);
    const int ncol  = lane & 15;
#pragma unroll
    for (int v = 0; v < 8; ++v) {
        const int ma = mbase + v;        // M-tile 0
        const int mb = ma + 16;          // M-tile 1
        P[(size_t)ma * PRED + (n0 + 0 * 16 + ncol)] = acc00[v];
        P[(size_t)ma * PRED + (n0 + 1 * 16 + ncol)] = acc01[v];
        P[(size_t)ma * PRED + (n0 + 2 * 16 + ncol)] = acc02[v];
        P[(size_t)mb * PRED + (n0 + 0 * 16 + ncol)] = acc10[v];
        P[(size_t)mb * PRED + (n0 + 1 * 16 + ncol)] = acc11[v];
        P[(size_t)mb * PRED + (n0 + 2 * 16 + ncol)] = acc12[v];
    }
}

// ---------------------------------------------------------------------------
// Kernel 4: deterministic reduction of the K-split partials + bias, scattered
// to out[b, p, c] (m = b*7 + c).
// ---------------------------------------------------------------------------
__global__ void reduce_kernel(const float* __restrict__ partial,
                              const float* __restrict__ bias,
                              float* __restrict__ out) {
    const int idx = blockIdx.x * blockDim.x + threadIdx.x;  // [0, 1792*720)
    const int m = idx / PRED;
    const int p = idx - m * PRED;
    const int b = m / CH, c = m % CH;
    float v = partial[idx] + partial[(size_t)M_ROWS * PRED + idx] + bias[p];
    out[((size_t)b * PRED + p) * CH + c] = v;
}

// ---------------------------------------------------------------------------
extern "C" void kernel_launch(void* const* d_in, const int* in_sizes, int n_in,
                              void* d_out, int out_size, void* d_ws, size_t ws_size,
                              hipStream_t stream) {
    (void)in_sizes; (void)n_in; (void)out_size; (void)ws_size;

    const float* x    = (const float*)d_in[0];   // [256, 96, 7]
    const float* W    = (const float*)d_in[1];   // [720, 7328]
    const float* bias = (const float*)d_in[2];   // [720]
    float* out        = (float*)d_out;           // [256, 720, 7]

    // Workspace layout (all 256B aligned):
    //   [0, 64KB)                    : index table (29 KB used)
    //   [64KB, 64KB + 52.53MB)       : A matrix  [1792 x 7328] f32
    //   [.., .. + 10.32MB)           : K-split partials, 2 x [1792 x 720] f32
    uint32_t* table = (uint32_t*)d_ws;
    float* A        = (float*)((char*)d_ws + (64 * 1024));
    float* partial  = (float*)((char*)d_ws + (64 * 1024) + (size_t)M_ROWS * K_FEAT * sizeof(float));

    const int sblocks = (K_FEAT + 255) / 256;  // 29

    build_table_kernel<<<dim3(sblocks), dim3(256), 0, stream>>>(table);
    expand_feat_kernel<<<dim3(sblocks, M_ROWS), dim3(256), 0, stream>>>(x, table, A);
    wmma_gemm_kernel<<<dim3(PRED / (16 * NTILES), M_ROWS / (128 * MTILES), KSPLIT),
                       dim3(256), 0, stream>>>(A, W, partial);
    reduce_kernel<<<dim3((M_ROWS * PRED) / 256), dim3(256), 0, stream>>>(partial, bias, out);
}